// SigWassersteinMetric_57217554317458
// MI455X (gfx1250) — compile-verified
//
#include <hip/hip_runtime.h>
#include <math.h>

// ---------------------------------------------------------------------------
// Depth-4 path signature Wasserstein-style metric for MI455X (gfx1250).
// Level-4 (90% of MACs) computed as three K=time GEMMs on V_WMMA_F32_16X16X4_F32:
//   S4 = sum_s (S3(s-)+w3/4) (x) v   ->  (1000x10, K=t) GEMM  (folds w4 term)
//      + sum_s  S2(s-)       (x) w2  ->  (100x100, K=t) GEMM
//      + sum_s  S1(s-)       (x) w3  ->  (10x1000, K=t) GEMM
// Only the cheap S1/S2/S3 prefix recurrences stay on VALU.
// ---------------------------------------------------------------------------

typedef __attribute__((ext_vector_type(2))) float v2f;
typedef __attribute__((ext_vector_type(8))) float v8f;

#define Cc     10
#define C2     100
#define C3     1000
#define C4     10000
#define DTOT   11110          // C + C^2 + C^3 + C^4
#define TLEN   256
#define NSTEP  255
#define CH     8              // time-chunk (GEMM K per chunk, 2 WMMA k-steps)
#define NCHUNK 32             // 32*8 = 256 >= 255 (last step zero-padded)
#define C2P    112            // 100 padded to 7*16
#define C3P    1008           // 1000 padded to 63*16
#define NT_A   63             // 1000x10 GEMM row tiles
#define NT_B   49             // 100x100 GEMM tiles (7x7)
#define NT_C   63             // 10x1000 GEMM col tiles
#define NT_TOT 175
#define THREADS 512
#define TPW    11             // ceil(175 tiles / 16 waves)

__global__ void zero_acc_kernel(float* __restrict__ acc) {
    int i = blockIdx.x * blockDim.x + threadIdx.x;
    if (i < DTOT) acc[i] = 0.0f;
}

__global__ __launch_bounds__(THREADS, 1)
void sig_kernel(const float* __restrict__ orig,
                const float* __restrict__ gen,
                const int*   __restrict__ sidx,
                float*       __restrict__ gacc)
{
    // ---- LDS layout (~87 KB; <= 160 KB so 2 WGs / WGP on 320 KB LDS) ----
    __shared__ float dxb[NSTEP * Cc];          // per-path increments
    __shared__ float sS1[16];                  // prefix signature level 1
    __shared__ float sS2[C2P];                 // prefix level 2
    __shared__ float sS3[C3P];                 // prefix level 3
    __shared__ float vbuf[CH][16];             // v_s           (B op of GEMM A)
    __shared__ float s1snap[CH][16];           // S1(s-)        (A op of GEMM C)
    __shared__ float w2buf[CH][C2P];           // v^2/2         (B op of GEMM B)
    __shared__ float s2snap[CH][C2P];          // S2(s-)        (A op of GEMM B)
    __shared__ float big[2 * CH * C3P];        // [0..CH): a_s = S3(s-)+w3/4 (A op GEMM A)
                                               // [CH..2CH): w3 = v^3/6      (B op GEMM C)
    float* s4 = big;                           // aliased 10000-float staging at the end

    const int tid  = threadIdx.x;
    const int wave = __builtin_amdgcn_readfirstlane((int)(threadIdx.x >> 5)); // SGPR -> scalar branches
    const int lane = tid & 31;
    const int lo   = lane & 15;                // N / M-row lane index
    const int hi   = lane >> 4;                // lane-half selects K pair / M+8

    // ---- path selection: +1 for selected originals, -1 for generated ----
    const int p = blockIdx.x;
    const float sign = (p < 1024) ? 1.0f : -1.0f;
    const float* __restrict__ src =
        (p < 1024) ? (orig + (size_t)sidx[p] * (TLEN * Cc))
                   : (gen  + (size_t)(p - 1024) * (TLEN * Cc));

    // ---- load path, form increments dx[s][c] = x[s+1][c] - x[s][c] ----
    for (int i = tid; i < NSTEP * Cc; i += THREADS)
        dxb[i] = src[i + Cc] - src[i];

    // ---- zero sequential state ----
    if (tid < 16)  sS1[tid] = 0.0f;
    if (tid < C2P) sS2[tid] = 0.0f;
    for (int i = tid; i < C3P; i += THREADS) sS3[i] = 0.0f;
    __syncthreads();

    // ---- per-wave WMMA accumulators, persist across all chunks ----
    v8f acc[TPW] = {};

    for (int ch = 0; ch < NCHUNK; ++ch) {
        // ---------------- sequential phase: build chunk operand buffers ----
        for (int ks = 0; ks < CH; ++ks) {
            const int s = ch * CH + ks;
            // phase 1 (merged): snapshot S1/S2, fetch v, and w2 = v(x)v/2
            // (w2 reads dxb directly -- no dependency on vbuf, saves a barrier)
            if (tid < C2) {
                s2snap[ks][tid] = sS2[tid];
                float w2v = 0.0f;
                if (s < NSTEP)
                    w2v = 0.5f * dxb[s * Cc + tid / 10] * dxb[s * Cc + tid % 10];
                w2buf[ks][tid] = w2v;
            }
            if (tid < Cc) {
                float vv = (s < NSTEP) ? dxb[s * Cc + tid] : 0.0f;
                vbuf[ks][tid]   = vv;
                s1snap[ks][tid] = sS1[tid];
            }
            __syncthreads();
            // phase 2: w3, a_s, sequential S3/S2/S1 updates
            for (int m = tid; m < C3; m += THREADS) {
                const int ij = m / 10, k = m % 10, i = m / 100, jk = m % 100;
                const float w2ij = w2buf[ks][ij];
                const float vk   = vbuf[ks][k];
                const float w3m  = (1.0f / 3.0f) * w2ij * vk;   // v^3/6 element
                big[(CH + ks) * C3P + m] = w3m;
                const float s3o = sS3[m];
                big[ks * C3P + m] = s3o + 0.25f * w3m;          // a_s (folds w4 = w3 (x) v / 4)
                // S3 += S2(s-) (x) v + S1(s-) (x) w2 + w3   (first+third fused)
                sS3[m] = s3o + (s2snap[ks][ij] + (1.0f / 3.0f) * w2ij) * vk
                             + s1snap[ks][i] * w2buf[ks][jk];
            }
            if (tid < C2)   // S2 += S1(s-) (x) v + w2   (reads snapshots only)
                sS2[tid] += (s1snap[ks][tid / 10] + 0.5f * vbuf[ks][tid / 10]) * vbuf[ks][tid % 10];
            if (tid < Cc)   // S1 += v
                sS1[tid] += vbuf[ks][tid];
            __syncthreads();
        }

        // ---------------- WMMA phase: 175 tiles over 16 waves, K = CH -------
        // ISA f32 16x16x4 A layout: VGPR0 = K{0,2}, VGPR1 = K{1,3} across lane halves;
        // B symmetric (row K striped over lanes-lo = N).  kb = kk + 2*hi.
        #pragma unroll
        for (int tl = 0; tl < TPW; ++tl) {
            const int t = wave * TPW + tl;
            if (t >= NT_TOT) break;
            #pragma unroll
            for (int kk = 0; kk < CH; kk += 4) {
                const int kb = kk + 2 * hi;
                v2f a, b;
                if (t < NT_A) {                      // GEMM A: a_s^T (1000xK) * v (Kx10)
                    const int m = t * 16 + lo;
                    a.x = big[kb * C3P + m];       a.y = big[(kb + 1) * C3P + m];
                    b.x = vbuf[kb][lo];            b.y = vbuf[kb + 1][lo];
                } else if (t < NT_A + NT_B) {        // GEMM B: S2snap^T (100xK) * w2 (Kx100)
                    const int u = t - NT_A, br = u / 7, bc = u % 7;
                    const int m = br * 16 + lo, nn = bc * 16 + lo;
                    a.x = s2snap[kb][m];           a.y = s2snap[kb + 1][m];
                    b.x = w2buf[kb][nn];           b.y = w2buf[kb + 1][nn];
                } else {                             // GEMM C: S1snap^T (10xK) * w3 (Kx1000)
                    const int ct = t - (NT_A + NT_B);
                    const int nn = ct * 16 + lo;
                    a.x = s1snap[kb][lo];          a.y = s1snap[kb + 1][lo];
                    b.x = big[(CH + kb) * C3P + nn]; b.y = big[(CH + kb + 1) * C3P + nn];
                }
                acc[tl] = __builtin_amdgcn_wmma_f32_16x16x4_f32(
                    false, a, false, b, (short)0, acc[tl], false, false);
            }
        }
        __syncthreads();   // buffers reused by next chunk
    }

    // ---------------- merge the three GEMM outputs into S4 staging ---------
    // C/D layout: VGPR r -> row (r + 8*hi), column lo.
    // Phase A (exclusive store, covers all of s4[0..10000)):
    #pragma unroll
    for (int tl = 0; tl < TPW; ++tl) {
        const int t = wave * TPW + tl;
        if (t >= NT_TOT) break;
        if (t < NT_A) {
            #pragma unroll
            for (int r = 0; r < 8; ++r) {
                const int m = t * 16 + r + hi * 8;
                if (m < C3 && lo < Cc) s4[m * Cc + lo] = acc[tl][r];
            }
        }
    }
    __syncthreads();
    #pragma unroll
    for (int tl = 0; tl < TPW; ++tl) {
        const int t = wave * TPW + tl;
        if (t >= NT_TOT) break;
        if (t >= NT_A && t < NT_A + NT_B) {
            const int u = t - NT_A, br = u / 7, bc = u % 7;
            #pragma unroll
            for (int r = 0; r < 8; ++r) {
                const int mg = br * 16 + r + hi * 8;
                const int ng = bc * 16 + lo;
                if (mg < C2 && ng < C2) s4[mg * C2 + ng] += acc[tl][r];
            }
        }
    }
    __syncthreads();
    #pragma unroll
    for (int tl = 0; tl < TPW; ++tl) {
        const int t = wave * TPW + tl;
        if (t >= NT_TOT) break;
        if (t >= NT_A + NT_B) {
            const int ct = t - (NT_A + NT_B);
            #pragma unroll
            for (int r = 0; r < 8; ++r) {
                const int mg = r + hi * 8;
                const int ng = ct * 16 + lo;
                if (mg < Cc && ng < C3) s4[mg * C3 + ng] += acc[tl][r];
            }
        }
    }
    __syncthreads();

    // ---------------- signed accumulation into global mean-diff vector -----
    for (int d = tid; d < DTOT; d += THREADS) {
        float val;
        if      (d < Cc)            val = sS1[d];
        else if (d < Cc + C2)       val = sS2[d - Cc];
        else if (d < Cc + C2 + C3)  val = sS3[d - (Cc + C2)];
        else                        val = s4[d - (Cc + C2 + C3)];
        atomicAdd(&gacc[d], sign * val);
    }
}

__global__ void norm_kernel(const float* __restrict__ gacc, float* __restrict__ out) {
    __shared__ float red[THREADS];
    float s = 0.0f;
    for (int d = threadIdx.x; d < DTOT; d += THREADS) {
        const float v = gacc[d];
        s += v * v;
    }
    red[threadIdx.x] = s;
    __syncthreads();
    for (int off = THREADS / 2; off > 0; off >>= 1) {
        if (threadIdx.x < (unsigned)off) red[threadIdx.x] += red[threadIdx.x + off];
        __syncthreads();
    }
    if (threadIdx.x == 0) out[0] = sqrtf(red[0]) * (1.0f / 1024.0f);  // / B, SCALE = 1
}

extern "C" void kernel_launch(void* const* d_in, const int* in_sizes, int n_in,
                              void* d_out, int out_size, void* d_ws, size_t ws_size,
                              hipStream_t stream) {
    (void)in_sizes; (void)n_in; (void)out_size; (void)ws_size;
    const float* orig = (const float*)d_in[0];   // (4096, 256, 10) f32
    const float* gen  = (const float*)d_in[1];   // (1024, 256, 10) f32
    const int*   sidx = (const int*)d_in[2];     // (1024,) int
    float* gacc = (float*)d_ws;                  // DTOT floats of scratch
    float* out  = (float*)d_out;

    zero_acc_kernel<<<(DTOT + 511) / 512, 512, 0, stream>>>(gacc);
    sig_kernel<<<2048, THREADS, 0, stream>>>(orig, gen, sidx, gacc);
    norm_kernel<<<1, THREADS, 0, stream>>>(gacc, out);
}